// ManifoldHyperConnections_14886356648849
// MI455X (gfx1250) — compile-verified
//
#include <hip/hip_runtime.h>
#include <hip/hip_bf16.h>
#include <stdint.h>

typedef __attribute__((ext_vector_type(2))) float v2f;
typedef __attribute__((ext_vector_type(8))) float v8f;

#define T_DIM   4096
#define HC_DIM  4
#define HID_DIM 4096
#define KDIM    (HC_DIM * HID_DIM)   // 16384
#define OUTD    24
#define NPAD    32
#define NITER   400                  // MAX_IT * MAX_IT
#define EPSV    1e-12f

// ---- GEMM tiling ----
#define KSPLIT  8                    // K-dimension split across blocks
#define KS      (KDIM / KSPLIT)      // 2048 K per block
#define MROWS   128                  // rows per block (8 waves x 16-row WMMA tiles)
#define KC      32                   // K columns per LDS chunk
#define NCHUNK  (KS / KC)            // 64 chunks
#define XSTRIDE (KC + 4)             // 36 floats: pad rows -> conflict-free ds_load_b64
#define BCHUNK  ((KC / 4) * 2 * 32 * 2) // 1024 floats of packed B per chunk
#define BP_FLOATS ((KDIM / 4) * 2 * 32 * 2) // 524288 floats (2 MB) packed B

// ---- gfx1250 async global->LDS path (probe via __has_builtin) ----
#if defined(__has_builtin)
# if __has_builtin(__builtin_amdgcn_global_load_async_to_lds_b128) && \
     __has_builtin(__builtin_amdgcn_s_wait_asynccnt)
#  define USE_ASYNC 1
# endif
#endif
#ifndef USE_ASYNC
# define USE_ASYNC 0
#endif

#if USE_ASYNC
// Builtin signature (from clang diagnostic): first param is
//   int __attribute__((vector_size(16))) __device__ *   (global, AS1, non-const)
// second param is the LDS-side pointer (AS3).
typedef int v4i_ __attribute__((vector_size(4 * sizeof(int))));
typedef __attribute__((address_space(1))) v4i_* gas128_t;
typedef __attribute__((address_space(3))) v4i_* las128_t;
#define GASYNC_B128(gp, lp)                                                  \
  __builtin_amdgcn_global_load_async_to_lds_b128(                            \
      (gas128_t)(uintptr_t)(gp),                                             \
      (las128_t)(uint32_t)(uintptr_t)(lp), 0, 0)
#endif

__device__ __forceinline__ v8f wmma_f32_4(v2f a, v2f b, v8f c) {
  // D = A(16x4) * B(4x16) + C, fp32, wave32
  return __builtin_amdgcn_wmma_f32_16x16x4_f32(false, a, false, b,
                                               (short)0, c, false, false);
}

// ------------------------------------------------------------------
// Kernel 0: zero C-partials + ssq workspace
// ------------------------------------------------------------------
__global__ void zero_ws_kernel(float* __restrict__ p, int n) {
  int i = blockIdx.x * blockDim.x + threadIdx.x;
  if (i < n) p[i] = 0.0f;
}

// ------------------------------------------------------------------
// Kernel 1: pack W (16384 x 24, pad N->32) into WMMA-B lane layout.
// Bpack float index: (kb*64 + nt*32 + lane)*2 + j
//   lane<16 : K = 4*kb+{0,1},  lane>=16 : K = 4*kb+{2,3};  n = nt*16 + (lane&15)
// ------------------------------------------------------------------
__global__ __launch_bounds__(256) void prepack_b_kernel(
    const float* __restrict__ W, float* __restrict__ Bp) {
  int idx = blockIdx.x * 256 + threadIdx.x;
  if (idx >= (KDIM / 4) * 64) return;
  int lane = idx & 31;
  int nt   = (idx >> 5) & 1;
  int kb   = idx >> 6;
  int koff = (lane >= 16) ? 2 : 0;
  int n    = nt * 16 + (lane & 15);
  int k0   = kb * 4 + koff;
  float b0 = (n < OUTD) ? W[(size_t)k0 * OUTD + n]       : 0.0f;
  float b1 = (n < OUTD) ? W[(size_t)(k0 + 1) * OUTD + n] : 0.0f;
  Bp[(size_t)idx * 2 + 0] = b0;
  Bp[(size_t)idx * 2 + 1] = b1;
}

// ------------------------------------------------------------------
// Kernel 2: partial GEMM + partial sum-of-squares.
// grid = 32 M-blocks * 8 K-splits; block = 256 threads (8 waves).
// Wave w owns rows [row0 + 16w, row0 + 16w + 16), accumulates over its
// K slice with V_WMMA_F32_16X16X4_F32, then atomically adds into cws.
// ------------------------------------------------------------------
__global__ __launch_bounds__(256) void gemm_partial_kernel(
    const float* __restrict__ x, const float* __restrict__ bpack,
    float* __restrict__ cws, float* __restrict__ ssqws) {
  __shared__ float xs[2][MROWS * XSTRIDE];  // 36 KB
  __shared__ float bs[2][BCHUNK];           // 8 KB

  const int tid  = threadIdx.x;
  const int wave = tid >> 5;
  const int lane = tid & 31;
  const int mblk = blockIdx.x / KSPLIT;
  const int kblk = blockIdx.x % KSPLIT;
  const int row0 = mblk * MROWS;
  const int k0   = kblk * KS;

  // loader mapping: 2 threads per row, 16 floats (4x float4) each
  const int lr = tid >> 1;
  const int lc = (tid & 1) * 16;
  const float* xsrc = x + (size_t)(row0 + lr) * KDIM + k0 + lc;

  auto issue_async = [&](int c) {
#if USE_ASYNC
    const float* gx = xsrc + c * KC;
    float* lx = &xs[c & 1][0] + lr * XSTRIDE + lc;
#pragma unroll
    for (int i = 0; i < 4; ++i) GASYNC_B128(gx + i * 4, lx + i * 4);
    const float* gb = bpack + (size_t)(k0 / 4 + c * (KC / 4)) * 128 + tid * 4;
    GASYNC_B128(gb, &bs[c & 1][0] + tid * 4);
#else
    (void)c;
#endif
  };

  float4 xr[4];
  float4 br;
  auto load_regs = [&](int c) {
    const float* gx = xsrc + c * KC;
#pragma unroll
    for (int i = 0; i < 4; ++i) xr[i] = *(const float4*)(gx + i * 4);
    br = *(const float4*)(bpack + (size_t)(k0 / 4 + c * (KC / 4)) * 128 + tid * 4);
  };
  auto store_regs = [&](int c) {
    float* lx = &xs[c & 1][0] + lr * XSTRIDE + lc;
#pragma unroll
    for (int i = 0; i < 4; ++i) *(float4*)(lx + i * 4) = xr[i];
    *(float4*)(&bs[c & 1][0] + tid * 4) = br;
  };

  v8f acc0 = {};
  v8f acc1 = {};
  float ssq = 0.0f;
  const int arow = (wave << 4) + (lane & 15);  // row within 128-row tile
  const int koff = (lane >> 4) << 1;           // 0 for lanes 0-15, 2 for 16-31

  auto compute = [&](int c) {
    const float* xb = &xs[c & 1][0] + arow * XSTRIDE + koff;
    const float* bb = &bs[c & 1][0] + lane * 2;
#pragma unroll
    for (int s = 0; s < KC / 4; ++s) {
      v2f a = *(const v2f*)(xb + s * 4);
      ssq += a.x * a.x + a.y * a.y;        // each x element counted once
      v2f b0 = *(const v2f*)(bb + (s * 2 + 0) * 64);
      v2f b1 = *(const v2f*)(bb + (s * 2 + 1) * 64);
      acc0 = wmma_f32_4(a, b0, acc0);
      acc1 = wmma_f32_4(a, b1, acc1);
    }
  };

#if USE_ASYNC
  issue_async(0);
  for (int c = 0; c < NCHUNK; ++c) {
    if (c + 1 < NCHUNK) {
      issue_async(c + 1);                        // prefetch next chunk
      __builtin_amdgcn_s_wait_asynccnt(5);       // chunk c complete
    } else {
      __builtin_amdgcn_s_wait_asynccnt(0);
    }
    __syncthreads();
    compute(c);
    __syncthreads();                             // safe to overwrite buf
  }
#else
  load_regs(0);
  for (int c = 0; c < NCHUNK; ++c) {
    store_regs(c);
    __syncthreads();
    if (c + 1 < NCHUNK) load_regs(c + 1);        // overlap with compute
    compute(c);
    __syncthreads();
  }
#endif

  // partial sum-of-squares: two lanes per row, KSPLIT blocks per row
  atomicAdd(&ssqws[row0 + arow], ssq);

  // C accumulators -> global partial sums.
  // acc element v at lane l = C[row = v + 8*(l>=16)][col = nt*16 + (l&15)]
  float* crow = cws + (size_t)(row0 + (wave << 4)) * NPAD;
  const int rhalf = (lane >> 4) << 3;
  const int ncol  = lane & 15;
#pragma unroll
  for (int v = 0; v < 8; ++v) {
    atomicAdd(&crow[(v + rhalf) * NPAD + ncol],      acc0[v]);
    atomicAdd(&crow[(v + rhalf) * NPAD + 16 + ncol], acc1[v]);
  }
}

// ------------------------------------------------------------------
// Kernel 3: epilogue — rms factor, sigmoids, exp, 400 Sinkhorn iters.
// One thread per row t.
// ------------------------------------------------------------------
__global__ __launch_bounds__(256) void epilogue_kernel(
    const float* __restrict__ cws, const float* __restrict__ ssqws,
    const float* __restrict__ bias, const float* __restrict__ alpha,
    float* __restrict__ out) {
  int t = blockIdx.x * blockDim.x + threadIdx.x;
  if (t >= T_DIM) return;

  float rinv = rsqrtf(ssqws[t] * (1.0f / (float)KDIM));
  const float* h = cws + (size_t)t * NPAD;
  float a0 = alpha[0], a1 = alpha[1], a2 = alpha[2];

#pragma unroll
  for (int c = 0; c < HC_DIM; ++c) {
    float z = rinv * a0 * h[c] + bias[c];
    out[(size_t)t * HC_DIM + c] = 1.0f / (1.0f + expf(-z));
  }
#pragma unroll
  for (int c = 0; c < HC_DIM; ++c) {
    float z = rinv * a1 * h[HC_DIM + c] + bias[HC_DIM + c];
    out[(size_t)T_DIM * HC_DIM + (size_t)t * HC_DIM + c] =
        2.0f / (1.0f + expf(-z));
  }

  float E[16];
#pragma unroll
  for (int i = 0; i < 16; ++i)
    E[i] = expf(rinv * a2 * h[2 * HC_DIM + i] + bias[2 * HC_DIM + i]);

  float u[4] = {1.f, 1.f, 1.f, 1.f};
  float v[4] = {1.f, 1.f, 1.f, 1.f};
  for (int it = 0; it < NITER; ++it) {
#pragma unroll
    for (int i = 0; i < 4; ++i) {
      float s = E[i * 4 + 0] * v[0] + E[i * 4 + 1] * v[1] +
                E[i * 4 + 2] * v[2] + E[i * 4 + 3] * v[3];
      u[i] = __builtin_amdgcn_rcpf(s + EPSV);
    }
#pragma unroll
    for (int j = 0; j < 4; ++j) {
      float s = E[0 * 4 + j] * u[0] + E[1 * 4 + j] * u[1] +
                E[2 * 4 + j] * u[2] + E[3 * 4 + j] * u[3];
      v[j] = __builtin_amdgcn_rcpf(s + EPSV);
    }
  }

  float* ores = out + (size_t)T_DIM * 8 + (size_t)t * 16;
#pragma unroll
  for (int i = 0; i < 4; ++i)
#pragma unroll
    for (int j = 0; j < 4; ++j)
      ores[i * 4 + j] = u[i] * E[i * 4 + j] * v[j];
}

// ------------------------------------------------------------------
extern "C" void kernel_launch(void* const* d_in, const int* in_sizes, int n_in,
                              void* d_out, int out_size, void* d_ws,
                              size_t ws_size, hipStream_t stream) {
  const float* x     = (const float*)d_in[0];  // (4096, 4, 4096) f32
  const float* w     = (const float*)d_in[1];  // (16384, 24) f32
  const float* bias  = (const float*)d_in[2];  // (24,) f32
  const float* alpha = (const float*)d_in[3];  // (3,) f32
  float* out = (float*)d_out;                  // 4096*24 f32 (concat tuple)

  float* bpack = (float*)d_ws;                 // 524288 floats (2 MB)
  float* cws   = bpack + BP_FLOATS;            // 4096*32 floats (512 KB)
  float* ssqws = cws + (size_t)T_DIM * NPAD;   // 4096 floats
  // total workspace used: ~2.64 MB

  int nz = T_DIM * NPAD + T_DIM;
  zero_ws_kernel<<<(nz + 255) / 256, 256, 0, stream>>>(cws, nz);
  prepack_b_kernel<<<((KDIM / 4) * 64 + 255) / 256, 256, 0, stream>>>(w, bpack);
  gemm_partial_kernel<<<(T_DIM / MROWS) * KSPLIT, 256, 0, stream>>>(
      x, bpack, cws, ssqws);
  epilogue_kernel<<<T_DIM / 256, 256, 0, stream>>>(cws, ssqws, bias, alpha, out);
}